// VectorQuantizer_27891517620807
// MI455X (gfx1250) — compile-verified
//
#include <hip/hip_runtime.h>
#include <math.h>

typedef float v2f __attribute__((ext_vector_type(2)));
typedef float v8f __attribute__((ext_vector_type(8)));

#define NROWS          131072      // 32*64*64 rows of dim 64
#define KCODES         1024
#define DIM            64
#define ROWS_PER_BLOCK 128         // 8 waves * 16 rows
#define CHUNK_CODES    128         // codebook rows staged in LDS per iteration
#define LDS_STRIDE     68          // 64 + 4 pad floats -> conflict-free b64 reads
#define NUMEL          8388608.0f

// ---------------------------------------------------------------------------
// Init: zero loss partials + histogram, precompute 0.5*||e_k||^2
// ---------------------------------------------------------------------------
__global__ __launch_bounds__(256) void vq_init(const float* __restrict__ E,
                                               float* __restrict__ partials,
                                               unsigned* __restrict__ counts,
                                               float* __restrict__ e2)
{
    int t = blockIdx.x * 256 + threadIdx.x;     // 40 blocks -> 10240 threads
    if (t < 8192) {
        partials[t] = 0.0f;
    } else if (t < 9216) {
        counts[t - 8192] = 0u;
    } else if (t < 10240) {
        int k = t - 9216;
        float s = 0.0f;
#pragma unroll
        for (int i = 0; i < DIM; ++i) { float e = E[k * DIM + i]; s += e * e; }
        e2[k] = 0.5f * s;
    }
}

// ---------------------------------------------------------------------------
// Main: WMMA f32 distance matmul + argmin + gather + partial loss + histogram
// ---------------------------------------------------------------------------
__global__ __launch_bounds__(256) void vq_main(const float* __restrict__ X,
                                               const float* __restrict__ E,
                                               const float* __restrict__ e2g,
                                               float* __restrict__ out,
                                               float* __restrict__ partials,
                                               unsigned* __restrict__ counts)
{
    __shared__ float e_lds[CHUNK_CODES * LDS_STRIDE];   // 34 KB, padded rows
    __shared__ float e2_lds[CHUNK_CODES];
    __shared__ int   idx_lds[8][16];

    const int tid  = threadIdx.x;
    const int w    = tid >> 5;          // wave id in block (wave32)
    const int lane = tid & 31;
    const int col  = lane & 15;         // N / M index within half
    const int half = lane >> 4;
    const int kh   = half << 1;         // K sub-offset per A/B f32 16x16x4 layout
    const int rowBase = blockIdx.x * ROWS_PER_BLOCK + w * 16;

    // Preload this wave's 16x64 X tile in A-operand layout:
    // lane(col,half): A.x = X[row=col][4j+kh], A.y = X[row=col][4j+kh+1]
    v2f xa[16];
#pragma unroll
    for (int j = 0; j < 16; ++j)
        xa[j] = *(const v2f*)(X + (rowBase + col) * DIM + j * 4 + kh);

    float bestVal[8];
    int   bestIdx[8];
#pragma unroll
    for (int r = 0; r < 8; ++r) { bestVal[r] = -3.402823466e38f; bestIdx[r] = 0; }

    for (int c = 0; c < KCODES / CHUNK_CODES; ++c) {
        // Cooperative load: 128 codebook rows (8192 floats) into padded LDS
        const float4* src = (const float4*)(E + c * CHUNK_CODES * DIM);
#pragma unroll
        for (int i = 0; i < 8; ++i) {
            int f   = tid + i * 256;          // float4 index within chunk
            int row = f >> 4;
            int kk  = (f & 15) << 2;
            *(float4*)(&e_lds[row * LDS_STRIDE + kk]) = src[f];
        }
        if (tid < CHUNK_CODES) e2_lds[tid] = e2g[c * CHUNK_CODES + tid];
        __syncthreads();

        // 8 subtiles of 16 codes; each: preload all 16 B operands from LDS,
        // then run the 16-deep WMMA accumulation chain (DS latency hidden
        // behind the chain via partial dscnt waits instead of per-pair drains).
        for (int s = 0; s < CHUNK_CODES / 16; ++s) {
            const float* eb = &e_lds[(s * 16 + col) * LDS_STRIDE + kh];
            const float  h  = e2_lds[s * 16 + col];     // issue early

            v2f bv[16];
#pragma unroll
            for (int j = 0; j < 16; ++j)
                bv[j] = *(const v2f*)(eb + j * 4);

            v8f acc = {0.f, 0.f, 0.f, 0.f, 0.f, 0.f, 0.f, 0.f};
#pragma unroll
            for (int j = 0; j < 16; ++j)
                acc = __builtin_amdgcn_wmma_f32_16x16x4_f32(
                        false, xa[j], false, bv[j], (short)0, acc, false, false);

            const int n = c * CHUNK_CODES + s * 16 + col;
            // score = x.e - 0.5||e||^2 ; maximize (== argmin distance).
            // Strictly-greater keeps earliest code on ties (argmin semantics).
#pragma unroll
            for (int r = 0; r < 8; ++r) {
                float sc = acc[r] - h;
                if (sc > bestVal[r]) { bestVal[r] = sc; bestIdx[r] = n; }
            }
        }
        __syncthreads();
    }

    // Cross-lane argmax over the 16 columns in each lane half
#pragma unroll
    for (int r = 0; r < 8; ++r) {
        float v  = bestVal[r];
        int   ix = bestIdx[r];
#pragma unroll
        for (int m = 1; m < 16; m <<= 1) {
            float ov = __shfl_xor(v, m, 32);
            int   oi = __shfl_xor(ix, m, 32);
            if (ov > v || (ov == v && oi < ix)) { v = ov; ix = oi; }
        }
        if (col == 0) {                       // lanes 0 (rows r) and 16 (rows 8+r)
            idx_lds[w][(half << 3) + r] = ix;
            atomicAdd(&counts[ix], 1u);       // integer atomics: deterministic
        }
    }
    __syncthreads();

    // Gather quantized rows, write output, accumulate (q-x)^2
    float lsum = 0.0f;
#pragma unroll
    for (int r = 0; r < 16; ++r) {
        int ix  = idx_lds[w][r];
        int row = rowBase + r;
        v2f q = *(const v2f*)(E + ix * DIM + lane * 2);
        v2f x = *(const v2f*)(X + row * DIM + lane * 2);
        *(v2f*)(out + row * DIM + lane * 2) = q;
        float dx = q.x - x.x, dy = q.y - x.y;
        lsum += dx * dx + dy * dy;
    }
#pragma unroll
    for (int m = 1; m < 32; m <<= 1) lsum += __shfl_xor(lsum, m, 32);
    if (lane == 0) partials[blockIdx.x * 8 + w] = lsum;   // fixed slot: deterministic
}

// ---------------------------------------------------------------------------
// Finalize: deterministic fixed-order reductions -> loss, perplexity
// ---------------------------------------------------------------------------
__global__ __launch_bounds__(1024) void vq_finalize(const float* __restrict__ partials,
                                                    const unsigned* __restrict__ counts,
                                                    float* __restrict__ out)
{
    __shared__ float red[1024];
    const int t = threadIdx.x;

    float s = 0.0f;
#pragma unroll
    for (int i = 0; i < 8; ++i) s += partials[t * 8 + i];
    red[t] = s;
    __syncthreads();
    for (int st = 512; st > 0; st >>= 1) {
        if (t < st) red[t] += red[t + st];
        __syncthreads();
    }
    float sumsq = red[0];
    __syncthreads();

    float p = (float)counts[t] * (1.0f / 131072.0f);
    red[t] = p * logf(p + 1e-10f);
    __syncthreads();
    for (int st = 512; st > 0; st >>= 1) {
        if (t < st) red[t] += red[t + st];
        __syncthreads();
    }
    if (t == 0) {
        // q_latent_loss + 0.25*e_latent_loss, both numerically mean((q-x)^2)
        out[8388608] = 1.25f * sumsq * (1.0f / NUMEL);
        out[8388609] = expf(-red[0]);
    }
}

// ---------------------------------------------------------------------------
extern "C" void kernel_launch(void* const* d_in, const int* in_sizes, int n_in,
                              void* d_out, int out_size, void* d_ws, size_t ws_size,
                              hipStream_t stream)
{
    const float* X = (const float*)d_in[0];   // [32,64,64,64] f32
    const float* E = (const float*)d_in[1];   // [1024,64] f32
    float* out = (float*)d_out;               // 8388608 quantized + loss + perplexity

    float*    ws       = (float*)d_ws;        // 40 KB scratch layout:
    float*    partials = ws;                  // [8192] per-wave loss partials
    unsigned* counts   = (unsigned*)(ws + 8192); // [1024] histogram
    float*    e2       = ws + 9216;           // [1024] 0.5*||e||^2

    vq_init<<<40, 256, 0, stream>>>(E, partials, counts, e2);
    vq_main<<<1024, 256, 0, stream>>>(X, E, e2, out, partials, counts);
    vq_finalize<<<1, 1024, 0, stream>>>(partials, counts, out);

    (void)in_sizes; (void)n_in; (void)out_size; (void)ws_size;
}